// graphSAGELayer_51565377356363
// MI455X (gfx1250) — compile-verified
//
#include <hip/hip_runtime.h>
#include <math.h>

typedef __attribute__((ext_vector_type(2))) float v2f;
typedef __attribute__((ext_vector_type(8))) float v8f;

#define NNODES 100000
#define DEG    16
#define DIM    128
#define DE     16
#define MEANW  (DIM + DE)          // 144
#define KTOT   (2 * DIM + DE)      // 272
#define DOUT   128

// ---------------------------------------------------------------------------
// Kernel 1: per-node gather + mean over DEG=16 edges.
// One wave32 per node. Lane l owns feature dims [4l, 4l+4): each neighbor row
// (512 B) is fetched with a single coalesced global_load_b128 per edge, and
// lanes 0..3 fetch the 64 B edge-feature row the same way. Mean rows are
// 576 B (16B-aligned), so stores are b128 too.
// ---------------------------------------------------------------------------
__global__ __launch_bounds__(256)
void sage_gather_mean(const float* __restrict__ h,
                      const float* __restrict__ edge_table,
                      const int*  __restrict__ neighbor_idx,
                      const int*  __restrict__ edge_idx,
                      float* __restrict__ mean) {
    const int lane = threadIdx.x & 31;
    const int node = blockIdx.x * 8 + (threadIdx.x >> 5);   // 12500 blocks * 8 waves
    const long ebase = (long)node * DEG;

    float4 ah = make_float4(0.f, 0.f, 0.f, 0.f);
    float4 ae = make_float4(0.f, 0.f, 0.f, 0.f);

    #pragma unroll 4
    for (int e = 0; e < DEG; ++e) {
        const int nbr = neighbor_idx[ebase + e];
        const int eid = edge_idx[ebase + e];
        const float4 hv =
            *(const float4*)(h + (long)nbr * DIM + lane * 4);   // b128, coalesced
        ah.x += hv.x; ah.y += hv.y; ah.z += hv.z; ah.w += hv.w;
        if (lane < 4) {
            const float4 ev =
                *(const float4*)(edge_table + (long)eid * DE + lane * 4);
            ae.x += ev.x; ae.y += ev.y; ae.z += ev.z; ae.w += ev.w;
        }
    }
    const float inv = 1.0f / (float)DEG;
    float* __restrict__ mp = mean + (long)node * MEANW;
    float4 mh = make_float4(ah.x * inv, ah.y * inv, ah.z * inv, ah.w * inv);
    *(float4*)(mp + lane * 4) = mh;                             // b128 store
    if (lane < 4) {
        float4 me = make_float4(ae.x * inv, ae.y * inv, ae.z * inv, ae.w * inv);
        *(float4*)(mp + DIM + lane * 4) = me;
    }
}

// ---------------------------------------------------------------------------
// Kernel 2: out = elu([h | mean] @ W) via V_WMMA_F32_16X16X4_F32.
// Block = 256 threads = 8 waves. Block covers 80 rows x 128 cols:
//   - A tile (80 x 272 fp32) staged in LDS, stride 274 (even -> aligned b64
//     reads; 274 % 64 = 18 -> conflict-free across the 16 rows of a fragment).
//   - wave w owns output cols [16w, 16w+16); loops K in steps of 4, loading one
//     B fragment from W and issuing 5 WMMAs (one per 16-row tile), so W traffic
//     is amortized 5x. 100000 / 80 = 1250 blocks exactly -> EXEC all-1s.
// Fragment layouts per ISA 7.12.2 (32-bit A 16x4, 32-bit C/D 16x16).
// ---------------------------------------------------------------------------
#define RT 5                       // row tiles of 16 per block
#define ASTRIDE 274

__global__ __launch_bounds__(256)
void sage_gemm_elu(const float* __restrict__ h,
                   const float* __restrict__ mean,
                   const float* __restrict__ W,
                   float* __restrict__ out) {
    __shared__ float Asub[RT * 16][ASTRIDE];

    const int tid  = threadIdx.x;
    const int wave = tid >> 5;          // 0..7  -> N tile
    const int lane = tid & 31;
    const long row0 = (long)blockIdx.x * (RT * 16);

    // Stage A = [h | mean] tile into LDS (fully coalesced global reads).
    for (int i = tid; i < RT * 16 * DIM; i += 256) {
        const int r = i >> 7, c = i & (DIM - 1);
        Asub[r][c] = h[(row0 + r) * DIM + c];
    }
    for (int i = tid; i < RT * 16 * MEANW; i += 256) {
        const int r = i / MEANW, c = i - r * MEANW;
        Asub[r][DIM + c] = mean[(row0 + r) * MEANW + c];
    }
    __syncthreads();

    v8f acc[RT];
    #pragma unroll
    for (int rt = 0; rt < RT; ++rt)
        #pragma unroll
        for (int j = 0; j < 8; ++j) acc[rt][j] = 0.0f;

    const int m     = lane & 15;          // A/B/C minor index within tile
    const int khalf = (lane >> 4) * 2;    // K sub-offset held by this lane
    const int ncol  = wave * 16 + m;      // output column

    const float* __restrict__ arow[RT];
    #pragma unroll
    for (int rt = 0; rt < RT; ++rt)
        arow[rt] = &Asub[rt * 16 + m][khalf];
    const float* __restrict__ wp = W + (long)khalf * DOUT + ncol;

    for (int k = 0; k < KTOT; k += 4) {
        v2f b;
        b.x = wp[k * DOUT];               // W[k+khalf  ][ncol]
        b.y = wp[k * DOUT + DOUT];        // W[k+khalf+1][ncol]
        #pragma unroll
        for (int rt = 0; rt < RT; ++rt) {
            const float2 av = *(const float2*)(arow[rt] + k);  // aligned ds b64
            v2f a; a.x = av.x; a.y = av.y;
            acc[rt] = __builtin_amdgcn_wmma_f32_16x16x4_f32(
                false, a, false, b, (short)0, acc[rt], false, false);
        }
    }

    // C/D layout: VGPR r holds rows {r, r+8}; lanes 0-15 -> M=r, 16-31 -> M=r+8.
    const int mhi = (lane >> 4) * 8;
    #pragma unroll
    for (int rt = 0; rt < RT; ++rt) {
        #pragma unroll
        for (int r = 0; r < 8; ++r) {
            const long node = row0 + rt * 16 + r + mhi;
            float x = acc[rt][r];
            x = x > 0.0f ? x : expm1f(x);   // jax.nn.elu, alpha = 1
            out[node * DOUT + ncol] = x;
        }
    }
}

// ---------------------------------------------------------------------------
extern "C" void kernel_launch(void* const* d_in, const int* in_sizes, int n_in,
                              void* d_out, int out_size, void* d_ws, size_t ws_size,
                              hipStream_t stream) {
    const float* h            = (const float*)d_in[0];
    const float* edge_table   = (const float*)d_in[1];
    const float* W            = (const float*)d_in[2];
    const int*   neighbor_idx = (const int*)d_in[3];
    const int*   edge_idx     = (const int*)d_in[4];
    // d_in[5] segment_ids unused: sorted fixed-degree layout is exploited directly.

    float* mean = (float*)d_ws;            // N * 144 fp32 = 57.6 MB scratch
    float* out  = (float*)d_out;           // N * 128 fp32

    sage_gather_mean<<<NNODES / 8, 256, 0, stream>>>(h, edge_table,
                                                     neighbor_idx, edge_idx, mean);
    sage_gemm_elu<<<NNODES / (RT * 16), 256, 0, stream>>>(h, mean, W, out);
}